// PointCloudNormals_57930518889114
// MI455X (gfx1250) — compile-verified
//
#include <hip/hip_runtime.h>
#include <math.h>

typedef float v2f __attribute__((ext_vector_type(2)));
typedef float v8f __attribute__((ext_vector_type(8)));

#define TPB 256
#define KNN 32

__device__ __forceinline__ void pick_evec(float a00, float a01, float a02,
                                          float a11, float a12, float a22,
                                          float lam, float& vx, float& vy, float& vz)
{
    // rows of (A - lam*I); eigenvector = best-conditioned cross product of two rows
    float r0x = a00 - lam, r0y = a01,       r0z = a02;
    float r1x = a01,       r1y = a11 - lam, r1z = a12;
    float r2x = a02,       r2y = a12,       r2z = a22 - lam;

    float cAx = r0y * r1z - r0z * r1y;
    float cAy = r0z * r1x - r0x * r1z;
    float cAz = r0x * r1y - r0y * r1x;
    float cBx = r0y * r2z - r0z * r2y;
    float cBy = r0z * r2x - r0x * r2z;
    float cBz = r0x * r2y - r0y * r2x;
    float cCx = r1y * r2z - r1z * r2y;
    float cCy = r1z * r2x - r1x * r2z;
    float cCz = r1x * r2y - r1y * r2x;

    float nA = cAx * cAx + cAy * cAy + cAz * cAz;
    float nB = cBx * cBx + cBy * cBy + cBz * cBz;
    float nC = cCx * cCx + cCy * cCy + cCz * cCz;

    float bx = cAx, by = cAy, bz = cAz, bn = nA;
    if (nB > bn) { bx = cBx; by = cBy; bz = cBz; bn = nB; }
    if (nC > bn) { bx = cCx; by = cCy; bz = cCz; bn = nC; }
    float inv = rsqrtf(fmaxf(bn, 1e-30f));
    vx = bx * inv; vy = by * inv; vz = bz * inv;
}

__global__ __launch_bounds__(TPB)
void shot_lrf_kernel(const float* __restrict__ verts, float* __restrict__ out, int N)
{
    // per-wave 32x32 d2' tile, stride 33 for bank-conflict-free row scans
    __shared__ float tile[8][32 * 33];

    const int tid  = threadIdx.x;
    const int lane = tid & 31;
    const int wave = tid >> 5;
    const int l15  = lane & 15;

    const int gq = blockIdx.x * TPB + tid;   // global query id (batch-major)
    const int b  = gq / N;
    const int q  = gq - b * N;
    const float* vb = verts + (size_t)b * N * 3;

    // own query
    const float qx = vb[(size_t)q * 3 + 0];
    const float qy = vb[(size_t)q * 3 + 1];
    const float qz = vb[(size_t)q * 3 + 2];
    const float qn2 = qx * qx + qy * qy + qz * qz;

    // Wave's A operands (16x4 f32, K=3 padded to 4):
    // lanes 0-15 hold (x,y)=K0,K1 of row M=lane; lanes 16-31 hold (z,0)=K2,K3 of row M=lane-16
    const int qbase = q - lane;  // wave-aligned (blocks never straddle batches: N % 256 == 0)
    const float* qa0 = vb + (size_t)(qbase + l15) * 3;
    const float* qa1 = vb + (size_t)(qbase + 16 + l15) * 3;
    v2f A0, A1;
    if (lane < 16) { A0[0] = qa0[0]; A0[1] = qa0[1]; A1[0] = qa1[0]; A1[1] = qa1[1]; }
    else           { A0[0] = qa0[2]; A0[1] = 0.0f;   A1[0] = qa1[2]; A1[1] = 0.0f;   }

    // register-resident sorted top-32 (ascending), static indexing only
    float dk[KNN];
    int   ik[KNN];
#pragma unroll
    for (int j = 0; j < KNN; ++j) { dk[j] = 3.0e38f; ik[j] = 0; }

    float* my_tile = &tile[wave][0];

    for (int cb = 0; cb < N; cb += 32) {
        // candidate coords for columns l15 and l15+16
        const float* cA = vb + (size_t)(cb + l15) * 3;
        const float* cB = vb + (size_t)(cb + 16 + l15) * 3;
        const float c0x = cA[0], c0y = cA[1], c0z = cA[2];
        const float c1x = cB[0], c1y = cB[1], c1z = cB[2];
        const float cn0 = c0x * c0x + c0y * c0y + c0z * c0z;
        const float cn1 = c1x * c1x + c1y * c1y + c1z * c1z;

        v2f B0, B1;
        if (lane < 16) { B0[0] = c0x; B0[1] = c0y; B1[0] = c1x; B1[1] = c1y; }
        else           { B0[0] = c0z; B0[1] = 0.0f; B1[0] = c1z; B1[1] = 0.0f; }

        v8f z8 = {};
        // D = A x B : 16x16 dot-product tiles, K=4 (coords padded), f32 accumulate
        v8f D00 = __builtin_amdgcn_wmma_f32_16x16x4_f32(false, A0, false, B0, (short)0, z8, false, false);
        v8f D01 = __builtin_amdgcn_wmma_f32_16x16x4_f32(false, A0, false, B1, (short)0, z8, false, false);
        v8f D10 = __builtin_amdgcn_wmma_f32_16x16x4_f32(false, A1, false, B0, (short)0, z8, false, false);
        v8f D11 = __builtin_amdgcn_wmma_f32_16x16x4_f32(false, A1, false, B1, (short)0, z8, false, false);

        // C/D layout: VGPR r -> row M=r (lanes 0-15) or M=r+8 (lanes 16-31), col N=lane%16
        const int mb = (lane < 16) ? 0 : 8;
#pragma unroll
        for (int r = 0; r < 8; ++r) {
            my_tile[(mb + r) * 33 + l15]           = cn0 - 2.0f * D00[r];
            my_tile[(mb + r) * 33 + 16 + l15]      = cn1 - 2.0f * D01[r];
            my_tile[(16 + mb + r) * 33 + l15]      = cn0 - 2.0f * D10[r];
            my_tile[(16 + mb + r) * 33 + 16 + l15] = cn1 - 2.0f * D11[r];
        }
        __syncthreads();

        // scan my query's row; maintain sorted top-32
        for (int n2 = 0; n2 < 32; ++n2) {
            float d2full = qn2 + my_tile[lane * 33 + n2];
            float dist = sqrtf(fmaxf(d2full, 1e-12f));
            if (dist < dk[KNN - 1]) {
                dk[KNN - 1] = dist;
                ik[KNN - 1] = cb + n2;
#pragma unroll
                for (int j = KNN - 1; j > 0; --j) {
                    if (dk[j] < dk[j - 1]) {
                        float tf = dk[j]; dk[j] = dk[j - 1]; dk[j - 1] = tf;
                        int   ti = ik[j]; ik[j] = ik[j - 1]; ik[j - 1] = ti;
                    }
                }
            }
        }
        __syncthreads();
    }

    // ---- weighted covariance over the K neighbors ----
    const float rad = dk[KNN - 1];
    float c00 = 0.f, c01 = 0.f, c02 = 0.f, c11 = 0.f, c12 = 0.f, c22 = 0.f, wsum = 0.f;
    for (int k = 0; k < KNN; ++k) {
        const float* pp = vb + (size_t)ik[k] * 3;
        float nx = pp[0] - qx, ny = pp[1] - qy, nz = pp[2] - qz;
        float w = rad - dk[k];
        wsum += w;
        c00 += w * nx * nx; c01 += w * nx * ny; c02 += w * nx * nz;
        c11 += w * ny * ny; c12 += w * ny * nz; c22 += w * nz * nz;
    }
    const float invw = 1.0f / fmaxf(wsum, 1e-30f);
    c00 *= invw; c01 *= invw; c02 *= invw; c11 *= invw; c12 *= invw; c22 *= invw;

    // ---- analytic 3x3 symmetric eigensolve ----
    float xx, xy, xz, zx, zy, zz;
    const float qm = (c00 + c11 + c22) * (1.0f / 3.0f);
    const float p1 = c01 * c01 + c02 * c02 + c12 * c12;
    const float dd0 = c00 - qm, dd1 = c11 - qm, dd2 = c22 - qm;
    const float p2 = dd0 * dd0 + dd1 * dd1 + dd2 * dd2 + 2.0f * p1;
    if (p2 < 1e-24f) {
        xx = 1.f; xy = 0.f; xz = 0.f;
        zx = 0.f; zy = 0.f; zz = 1.f;
    } else {
        const float p = sqrtf(p2 * (1.0f / 6.0f));
        const float invp = 1.0f / p;
        const float b00 = dd0 * invp, b11 = dd1 * invp, b22 = dd2 * invp;
        const float b01 = c01 * invp, b02 = c02 * invp, b12 = c12 * invp;
        float detB = b00 * (b11 * b22 - b12 * b12)
                   - b01 * (b01 * b22 - b12 * b02)
                   + b02 * (b01 * b12 - b11 * b02);
        float rr = fminf(1.0f, fmaxf(-1.0f, 0.5f * detB));
        const float phi = acosf(rr) * (1.0f / 3.0f);
        const float lmax = qm + 2.0f * p * cosf(phi);                       // largest  -> x
        const float lmin = qm + 2.0f * p * cosf(phi + 2.0943951023931953f); // smallest -> z
        pick_evec(c00, c01, c02, c11, c12, c22, lmax, xx, xy, xz);
        pick_evec(c00, c01, c02, c11, c12, c22, lmin, zx, zy, zz);
    }

    // ---- sign disambiguation ----
    int npx = 0, npz = 0;
    for (int k = 0; k < KNN; ++k) {
        const float* pp = vb + (size_t)ik[k] * 3;
        float nx = pp[0] - qx, ny = pp[1] - qy, nz = pp[2] - qz;
        if (nx * xx + ny * xy + nz * xz >= 0.f) npx++;
        if (nx * zx + ny * zy + nz * zz >= 0.f) npz++;
    }
    if (npx < KNN - npx) { xx = -xx; xy = -xy; xz = -xz; }
    if (npz < KNN - npz) { zx = -zx; zy = -zy; zz = -zz; }

    // y = z cross x
    const float yx = zy * xz - zz * xy;
    const float yy = zz * xx - zx * xz;
    const float yz = zx * xy - zy * xx;

    float* o = out + (size_t)gq * 9;
    o[0] = xx; o[1] = xy; o[2] = xz;
    o[3] = yx; o[4] = yy; o[5] = yz;
    o[6] = zx; o[7] = zy; o[8] = zz;
}

extern "C" void kernel_launch(void* const* d_in, const int* in_sizes, int n_in,
                              void* d_out, int out_size, void* d_ws, size_t ws_size,
                              hipStream_t stream)
{
    const float* verts = (const float*)d_in[0];
    float* out = (float*)d_out;
    const int total = in_sizes[0];   // B * N * 3, B = 2
    const int N = total / 6;         // per-batch point count (8192)
    const int total_threads = 2 * N;
    const int blocks = (total_threads + TPB - 1) / TPB;
    shot_lrf_kernel<<<blocks, TPB, 0, stream>>>(verts, out, N);
}